// Wasserstein1DLoss_54881092108429
// MI455X (gfx1250) — compile-verified
//
#include <hip/hip_runtime.h>

// Wasserstein-1D loss vs one-hot target:
//   w1(row) = sum_j softmax(x)_j * |j - t| / C      (cumsum form collapses to this)
//   loss    = mean over rows
//
// Kernel 1: one block (8 wave32) per row. Async global->LDS stage (CDNA5
//           GLOBAL_LOAD_ASYNC_TO_LDS_B128 / ASYNCcnt), then register-resident
//           two-phase reduction (max, then sum e and sum e*|j-t|).
// Kernel 2: 8192 -> 1 reduction using V_WMMA_F32_16X16X4_F32 as a streaming
//           accumulator (B = ones, C accumulates row sums).

namespace {

constexpr int kClasses = 4096;
constexpr int kBatch   = 8192;
constexpr int kThreads = 256;   // 8 waves of 32
constexpr int kChunks  = 4;     // 4 x float4 per thread = 16 elems = 4096/256

typedef __attribute__((ext_vector_type(2))) float v2f;
typedef __attribute__((ext_vector_type(8))) float v8f;

__global__ __launch_bounds__(kThreads)
void w1_rows_kernel(const float* __restrict__ logits,
                    const int*  __restrict__ targets,
                    float* __restrict__ w1out)
{
    __shared__ __align__(16) float row[kClasses];   // 16 KB staging
    __shared__ float smax[8];
    __shared__ float snum[8];
    __shared__ float sden[8];

    const int tid  = threadIdx.x;
    const int lane = tid & 31;
    const int wid  = tid >> 5;
    const int r    = blockIdx.x;

    const float* src = logits + (size_t)r * kClasses;

    // ---- async global -> LDS stage (tracked by ASYNCcnt) ----
    #pragma unroll
    for (int k = 0; k < kChunks; ++k) {
        const int idx = k * (kThreads * 4) + tid * 4;           // element index
        // low 32 bits of a flat pointer into LDS == LDS byte offset
        unsigned ldsOff = (unsigned)(uintptr_t)(row + idx);
        unsigned long long gaddr = (unsigned long long)(uintptr_t)(src + idx);
        asm volatile("global_load_async_to_lds_b128 %0, %1, off"
                     :: "v"(ldsOff), "v"(gaddr) : "memory");
    }
    asm volatile("s_wait_asynccnt 0" ::: "memory");
    __syncthreads();

    // ---- pull this thread's 16 elements into registers (read LDS once) ----
    float4 v[kChunks];
    #pragma unroll
    for (int k = 0; k < kChunks; ++k)
        v[k] = *(const float4*)(&row[k * (kThreads * 4) + tid * 4]);

    // ---- pass 1: row max ----
    float m = -3.402823466e+38f;
    #pragma unroll
    for (int k = 0; k < kChunks; ++k)
        m = fmaxf(fmaxf(fmaxf(m, v[k].x), fmaxf(v[k].y, v[k].z)), v[k].w);
    #pragma unroll
    for (int o = 16; o > 0; o >>= 1) m = fmaxf(m, __shfl_xor(m, o, 32));
    if (lane == 0) smax[wid] = m;
    __syncthreads();
    float M = smax[0];
    #pragma unroll
    for (int i = 1; i < 8; ++i) M = fmaxf(M, smax[i]);

    // ---- pass 2: den = sum exp, num = sum exp * |j - t| ----
    const int t = targets[r];
    float num = 0.f, den = 0.f;
    #pragma unroll
    for (int k = 0; k < kChunks; ++k) {
        const int j0 = k * (kThreads * 4) + tid * 4;
        const float e0 = __expf(v[k].x - M);
        const float e1 = __expf(v[k].y - M);
        const float e2 = __expf(v[k].z - M);
        const float e3 = __expf(v[k].w - M);
        den += (e0 + e1) + (e2 + e3);
        num += e0 * fabsf((float)(j0     - t))
             + e1 * fabsf((float)(j0 + 1 - t))
             + e2 * fabsf((float)(j0 + 2 - t))
             + e3 * fabsf((float)(j0 + 3 - t));
    }
    #pragma unroll
    for (int o = 16; o > 0; o >>= 1) {
        num += __shfl_xor(num, o, 32);
        den += __shfl_xor(den, o, 32);
    }
    if (lane == 0) { snum[wid] = num; sden[wid] = den; }
    __syncthreads();
    if (tid == 0) {
        float N = 0.f, D = 0.f;
        #pragma unroll
        for (int i = 0; i < 8; ++i) { N += snum[i]; D += sden[i]; }
        w1out[r] = N / (D * (float)kClasses);
    }
}

// Final mean over 8192 per-row losses. Each wave reduces 1024 values by
// streaming 16x4 f32 tiles through V_WMMA_F32_16X16X4_F32 with B = ones and
// accumulating into C: every D column then carries all 16 row sums, so the
// wave total = sum of D[m, 0] over m (lane 0 vgprs 0..7 + lane 16 vgprs 0..7).
__global__ __launch_bounds__(kThreads)
void finalize_kernel(const float* __restrict__ w1, float* __restrict__ out)
{
    __shared__ float part[8];
    const int tid  = threadIdx.x;
    const int lane = tid & 31;
    const int wid  = tid >> 5;

    const float* base = w1 + wid * (kBatch / 8);   // 1024 values per wave

    v2f b; b.x = 1.0f; b.y = 1.0f;                 // B (4x16) = all ones
    v8f c = {};                                    // C/D accumulator
    #pragma unroll
    for (int i = 0; i < 16; ++i) {
        v2f a;
        a.x = base[i * 64 + lane];
        a.y = base[i * 64 + 32 + lane];
        // (neg_a, A, neg_b, B, c_mod, C, reuse_a, reuse_b)
        c = __builtin_amdgcn_wmma_f32_16x16x4_f32(false, a, false, b,
                                                  (short)0, c, false, false);
    }

    float s = 0.f;
    #pragma unroll
    for (int i = 0; i < 8; ++i)
        s += __shfl(c[i], 0, 32) + __shfl(c[i], 16, 32);

    if (lane == 0) part[wid] = s;
    __syncthreads();
    if (tid == 0) {
        float tot = 0.f;
        #pragma unroll
        for (int i = 0; i < 8; ++i) tot += part[i];
        out[0] = tot * (1.0f / (float)kBatch);   // SCALE_FACTOR == 1.0
    }
}

} // namespace

extern "C" void kernel_launch(void* const* d_in, const int* in_sizes, int n_in,
                              void* d_out, int out_size, void* d_ws, size_t ws_size,
                              hipStream_t stream) {
    const float* logits  = (const float*)d_in[0];   // (8192, 4096) f32
    const int*   targets = (const int*)d_in[1];     // (8192,) int
    float* ws  = (float*)d_ws;                      // 8192 per-row losses
    float* out = (float*)d_out;                     // scalar f32

    (void)in_sizes; (void)n_in; (void)out_size; (void)ws_size;

    w1_rows_kernel<<<kBatch, kThreads, 0, stream>>>(logits, targets, ws);
    finalize_kernel<<<1, kThreads, 0, stream>>>(ws, out);
}